// GRDPI_inv_Net_86870008529011
// MI455X (gfx1250) — compile-verified
//
#include <hip/hip_runtime.h>
#include <stdint.h>

// ---------------------------------------------------------------------------
// GRDPI hysteresis scan for MI455X (gfx1250).
//   x, x_pre : [B, T] float32 streams (B=4096, T=2048)
//   z state  : 4 floats per row, serial recurrence in T
// Memory-bound (≈96 MB @ 23.3 TB/s ≈ 4.1 us floor). Parallelism = B rows.
// Data path: TDM tensor_load_to_lds double-buffered 32x32 tiles with LDS
// padding for bank-conflict-free reads; s_wait_tensorcnt for dependencies.
// ---------------------------------------------------------------------------

typedef __attribute__((ext_vector_type(4))) unsigned int u32x4;
typedef __attribute__((ext_vector_type(8))) unsigned int u32x8;

constexpr int BATCH = 4096;
constexpr int TLEN  = 2048;
constexpr int CHUNK = 32;            // timesteps per staged tile
constexpr int NCH   = TLEN / CHUNK;  // 64 chunks
constexpr int ROWS  = 32;            // rows per block == wave32 (single-wave WG)
constexpr int RS    = 33;            // padded LDS row stride in dwords (33 coprime 64 banks)

#define WAIT_TENSORCNT(imm) asm volatile("s_wait_tensorcnt " imm ::: "memory")

__device__ __forceinline__ uint32_t lds_offset(const void* p) {
  // Generic pointer to LDS: addr[31:0] is the hardware LDS byte address.
  return (uint32_t)(uintptr_t)p;
}

// One TDM descriptor load: 32 rows x 32 dwords tile, row stride TLEN dwords in
// global, row stride RS dwords in LDS (pad_interval=32 dwords, pad_amount=1).
__device__ __forceinline__ void tdm_load_tile(uint32_t lds, const float* g) {
  const uint64_t ga = (uint64_t)(uintptr_t)g;

  u32x4 g0;
  g0[0] = 1u;                                                   // count=1, user load
  g0[1] = lds;                                                  // lds_addr (bytes)
  g0[2] = (uint32_t)ga;                                         // global_addr[31:0]
  g0[3] = ((uint32_t)(ga >> 32) & 0x01FFFFFFu) | (2u << 30);    // addr[56:32] | type=2

  u32x8 g1;
  g1[0] = (2u << 16)      // data_size = 4 bytes
        | (1u << 20)      // pad_enable
        | (4u << 22)      // pad_interval: 2^(4+1) = 32 dwords
        | (0u << 25);     // pad_amount: 1 dword  -> LDS row stride = 33 dwords
  g1[1] = 0xFFFF0000u;    // barrier_addr=0 | tensor_dim0[15:0]<<16 (dim0=0x7FFFFFFF)
  g1[2] = 0xFFFF7FFFu;    // tensor_dim0[31:16] | tensor_dim1[15:0]<<16 (dim1=0x7FFFFFFF)
  g1[3] = 0x00207FFFu;    // tensor_dim1[31:16] | tile_dim0(32)<<16
  g1[4] = 32u;            // tile_dim1 = 32, tile_dim2 = 0 (2D)
  g1[5] = (uint32_t)TLEN; // tensor_dim0_stride[31:0] = 2048 elements
  g1[6] = 0u;             // stride0[47:32] | tensor_dim1_stride[15:0] (unused, 2D)
  g1[7] = 0u;

  u32x4 g2 = {0u, 0u, 0u, 0u};   // dims 2/3 unused (tile_dim2 = 0)
  u32x4 g3 = {0u, 0u, 0u, 0u};

  asm volatile("tensor_load_to_lds %0, %1, %2, %3"
               :: "s"(g0), "s"(g1), "s"(g2), "s"(g3)
               : "memory");
}

__global__ __launch_bounds__(ROWS) void grdpi_scan_kernel(
    const float* __restrict__ x, const float* __restrict__ xpre,
    const float* __restrict__ zinit, const float* __restrict__ w,
    float* __restrict__ out, float* __restrict__ zfin)
{
  __shared__ float xs[2][ROWS * RS];   // double-buffered x tiles
  __shared__ float ps[2][ROWS * RS];   // double-buffered x_pre tiles

  const int lane = threadIdx.x;
  const int b    = blockIdx.x * ROWS + lane;

  // ---- weights -> scan coefficients (wave-uniform) ----
  const float a0 = w[0], a1 = w[1], a2 = w[2], a3 = w[3], a4 = w[4];
  const float h  = w[5], beta = w[6], bb = w[7];
  const float EPS = 1e-10f;

  float ch[4];
  ch[0] = a0 * h;
  ch[1] = (2.f * a0 + a1) * h;
  ch[2] = (3.f * a0 + 2.f * a1 + a2) * h;
  ch[3] = (4.f * a0 + 3.f * a1 + 2.f * a2 + a3) * h;
  const float a0beta = a0 * beta;

  const float arest[4] = {a1, a2, a3, a4};
  float cum = a0;
  float inv_prev = 1.f / (cum + EPS);
  const float b0 = inv_prev;
  float bc[4];
#pragma unroll
  for (int i = 0; i < 4; ++i) {
    cum += arest[i];
    const float inv = 1.f / (cum + EPS);
    bc[i] = inv - inv_prev;
    inv_prev = inv;
  }

  // ---- per-row state ----
  float z[4];
#pragma unroll
  for (int i = 0; i < 4; ++i) z[i] = zinit[b * 4 + i];

  const float* gx = x    + (size_t)blockIdx.x * ROWS * TLEN;  // block tile base
  const float* gp = xpre + (size_t)blockIdx.x * ROWS * TLEN;
  float*       go = out  + (size_t)b * TLEN;                  // this lane's row

  const uint32_t lx[2] = { lds_offset(&xs[0][0]), lds_offset(&xs[1][0]) };
  const uint32_t lp[2] = { lds_offset(&ps[0][0]), lds_offset(&ps[1][0]) };

  // prologue: stage chunk 0
  tdm_load_tile(lx[0], gx);
  tdm_load_tile(lp[0], gp);

  for (int c = 0; c < NCH; ++c) {
    const int buf = c & 1;
    if (c + 1 < NCH) {
      // prefetch next chunk into the other buffer, then wait for chunk c
      tdm_load_tile(lx[buf ^ 1], gx + (c + 1) * CHUNK);
      tdm_load_tile(lp[buf ^ 1], gp + (c + 1) * CHUNK);
      WAIT_TENSORCNT("0x2");   // TDM ops complete in order: chunk c's 2 are done
    } else {
      WAIT_TENSORCNT("0x0");
    }

    const float* xr = &xs[buf][lane * RS];   // stride 33 dwords: conflict-free
    const float* pr = &ps[buf][lane * RS];

    float o[CHUNK];
#pragma unroll
    for (int t = 0; t < CHUNK; ++t) {
      const float u  = xr[t];
      const float up = pr[t];
      const float du = u - up;
      const float sq = sqrtf(fmaf(du, du, EPS));
      float acc = fmaf(b0, u, bb);
#pragma unroll
      for (int i = 0; i < 4; ++i) {
        const float r  = fmaf(a0beta, sq, ch[i]);          // threshold s_i
        const float hi = u + r;
        const float lo = u - r;
        // stop operator, written exactly as the reference relu chain
        const float inner = fmaxf(hi - z[i], 0.f);
        z[i] = lo + fmaxf(hi - inner - lo, 0.f);
        acc = fmaf(bc[i], z[i], acc);
      }
      o[t] = acc;
    }

    // 8x b128 stores per lane; (b*T + c*32)*4 is 16B aligned
#pragma unroll
    for (int j = 0; j < CHUNK / 4; ++j) {
      float4 v = make_float4(o[4 * j], o[4 * j + 1], o[4 * j + 2], o[4 * j + 3]);
      *reinterpret_cast<float4*>(go + c * CHUNK + 4 * j) = v;
    }
  }

#pragma unroll
  for (int i = 0; i < 4; ++i) zfin[b * 4 + i] = z[i];
}

// ---------------------------------------------------------------------------
// d_in: [0]=x (B*T f32), [1]=x_pre (B*T f32), [2]=z_init (B*4 f32), [3]=weights (11 f32)
// d_out: [B*T custom_output | 11 weights | B*4 z_final] f32
// ---------------------------------------------------------------------------
extern "C" void kernel_launch(void* const* d_in, const int* in_sizes, int n_in,
                              void* d_out, int out_size, void* d_ws, size_t ws_size,
                              hipStream_t stream) {
  (void)in_sizes; (void)n_in; (void)out_size; (void)d_ws; (void)ws_size;

  const float* x     = (const float*)d_in[0];
  const float* xpre  = (const float*)d_in[1];
  const float* zinit = (const float*)d_in[2];
  const float* w     = (const float*)d_in[3];

  float* out  = (float*)d_out;
  float* wout = out + (size_t)BATCH * TLEN;   // weights passthrough
  float* zfin = wout + 11;                    // z_final

  hipMemcpyAsync(wout, w, 11 * sizeof(float), hipMemcpyDeviceToDevice, stream);

  grdpi_scan_kernel<<<BATCH / ROWS, ROWS, 0, stream>>>(x, xpre, zinit, w, out, zfin);
}